// LightGCN_37211596652874
// MI455X (gfx1250) — compile-verified
//
#include <hip/hip_runtime.h>
#include <cstdint>
#include <cstddef>

// ---------------------------------------------------------------------------
// LightGCN 3-hop propagation for MI455X (gfx1250).
// CSR built once per call (deterministic). Hop SpMM runs as one-hot-expansion
// WMMA: per 16-row block, 4 edges/step -> A[16x4] one-hot values, B[4x16]
// gathered embedding tile, C[16x16] accumulated with V_WMMA_F32_16X16X4_F32.
// Edge tiles staged to LDS with CDNA5 GLOBAL_LOAD_ASYNC_TO_LDS + ASYNCcnt.
// JAX threefry2x32 replicated exactly for both dropout masks.
// ---------------------------------------------------------------------------

constexpr int N_USERS    = 100000;
constexpr int N_ITEMS    = 200000;
constexpr int N_NODES    = N_USERS + N_ITEMS;     // 300000
constexpr int EMB        = 64;
constexpr int N_HOPS     = 3;
constexpr int NNZ        = 2000000;
constexpr int HALF_E     = NNZ / 2;               // 1,000,000
constexpr int MSG_N      = N_NODES * EMB;         // 19,200,000
constexpr int HALF_M     = MSG_N / 2;             // 9,600,000
constexpr int OUT_STRIDE = N_HOPS * EMB;          // 192
constexpr int SCAN_B     = (N_NODES + 1023) / 1024; // 293

typedef __attribute__((ext_vector_type(2))) float v2f;
typedef __attribute__((ext_vector_type(8))) float v8f;

#define DEVHOST __host__ __device__

DEVHOST inline uint32_t rotl32(uint32_t x, int r) { return (x << r) | (x >> (32 - r)); }

// Exact JAX threefry2x32 (20 rounds).
DEVHOST inline void threefry2x32(uint32_t k0, uint32_t k1, uint32_t x0, uint32_t x1,
                                 uint32_t& o0, uint32_t& o1) {
  const uint32_t ks2 = k0 ^ k1 ^ 0x1BD11BDAu;
  x0 += k0; x1 += k1;
#define TFR(r) { x0 += x1; x1 = rotl32(x1, r); x1 ^= x0; }
  TFR(13) TFR(15) TFR(26) TFR(6)   x0 += k1;  x1 += ks2 + 1u;
  TFR(17) TFR(29) TFR(16) TFR(24)  x0 += ks2; x1 += k0  + 2u;
  TFR(13) TFR(15) TFR(26) TFR(6)   x0 += k0;  x1 += k1  + 3u;
  TFR(17) TFR(29) TFR(16) TFR(24)  x0 += k1;  x1 += ks2 + 4u;
  TFR(13) TFR(15) TFR(26) TFR(6)   x0 += ks2; x1 += k0  + 5u;
#undef TFR
  o0 = x0; o1 = x1;
}

__device__ inline float bits_to_unit(uint32_t b) {
  union { uint32_t u; float f; } c;
  c.u = (b >> 9) | 0x3f800000u;
  return c.f - 1.0f;   // U[0,1), exactly as jax.random.uniform
}

// ---------------------------------------------------------------------------
// Utility kernels
// ---------------------------------------------------------------------------
__global__ __launch_bounds__(256) void zero_i32(int* __restrict__ p, int n) {
  int i = blockIdx.x * 256 + threadIdx.x;
  if (i < n) p[i] = 0;
}

__global__ __launch_bounds__(256) void zero_f4(float4* __restrict__ p, int n4) {
  int i = blockIdx.x * 256 + threadIdx.x;
  if (i < n4) p[i] = make_float4(0.f, 0.f, 0.f, 0.f);
}

// ---------------------------------------------------------------------------
// CSR build (once per call; deterministic after per-row sort by edge index)
// ---------------------------------------------------------------------------
__global__ __launch_bounds__(256) void hist_rows(const int* __restrict__ rows,
                                                 int* __restrict__ cnt) {
  int e = blockIdx.x * 256 + threadIdx.x;
  if (e < NNZ) atomicAdd(&cnt[rows[e]], 1);
}

__global__ __launch_bounds__(256) void scan_k1(const int* __restrict__ cnt,
                                               int* __restrict__ bsum) {
  __shared__ int s[256];
  int b = blockIdx.x, t = threadIdx.x;
  int base = b * 1024 + t * 4, sum = 0;
  for (int k = 0; k < 4; ++k) { int i = base + k; if (i < N_NODES) sum += cnt[i]; }
  s[t] = sum;
  for (int off = 128; off > 0; off >>= 1) {
    __syncthreads();
    if (t < off) s[t] += s[t + off];
  }
  if (t == 0) bsum[b] = s[0];
}

__global__ void scan_k2(int* __restrict__ bsum, int* __restrict__ rowptr) {
  if (threadIdx.x == 0 && blockIdx.x == 0) {
    int run = 0;
    for (int b = 0; b < SCAN_B; ++b) { int x = bsum[b]; bsum[b] = run; run += x; }
    rowptr[N_NODES] = run;   // == NNZ
  }
}

__global__ __launch_bounds__(256) void scan_k3(const int* __restrict__ cnt,
                                               const int* __restrict__ bsum,
                                               int* __restrict__ rowptr,
                                               int* __restrict__ cursor) {
  __shared__ int s[256];
  int b = blockIdx.x, t = threadIdx.x;
  int base = b * 1024 + t * 4;
  int v[4]; int sum = 0;
  for (int k = 0; k < 4; ++k) { int i = base + k; v[k] = (i < N_NODES) ? cnt[i] : 0; sum += v[k]; }
  s[t] = sum; __syncthreads();
  for (int off = 1; off < 256; off <<= 1) {
    int x = (t >= off) ? s[t - off] : 0;
    __syncthreads();
    s[t] += x;
    __syncthreads();
  }
  int run = bsum[b] + (s[t] - sum);
  for (int k = 0; k < 4; ++k) {
    int i = base + k;
    if (i < N_NODES) { rowptr[i] = run; cursor[i] = run; }
    run += v[k];
  }
}

__global__ __launch_bounds__(256) void scatter_csr(const int* __restrict__ rows,
                                                   const int* __restrict__ cols,
                                                   const float* __restrict__ vals,
                                                   int* __restrict__ cursor,
                                                   int* __restrict__ eidx_s,
                                                   int* __restrict__ col_s,
                                                   float* __restrict__ val_s) {
  int e = blockIdx.x * 256 + threadIdx.x;
  if (e >= NNZ) return;
  int r = rows[e];
  int pos = atomicAdd(&cursor[r], 1);
  eidx_s[pos] = e;
  col_s[pos]  = cols[e];
  val_s[pos]  = vals[e];
}

// Per-row insertion sort by original edge index -> deterministic CSR contents
// and summation order matching segment_sum's edge order.
__global__ __launch_bounds__(256) void sort_rows(const int* __restrict__ rowptr,
                                                 int* __restrict__ eidx_s,
                                                 int* __restrict__ col_s,
                                                 float* __restrict__ val_s) {
  int r = blockIdx.x * 256 + threadIdx.x;
  if (r >= N_NODES) return;
  int jb = rowptr[r], je = rowptr[r + 1];
  for (int i = jb + 1; i < je; ++i) {
    int ke = eidx_s[i]; int kc = col_s[i]; float kv = val_s[i];
    int j = i - 1;
    while (j >= jb && eidx_s[j] > ke) {
      eidx_s[j + 1] = eidx_s[j]; col_s[j + 1] = col_s[j]; val_s[j + 1] = val_s[j];
      --j;
    }
    eidx_s[j + 1] = ke; col_s[j + 1] = kc; val_s[j + 1] = kv;
  }
}

// ---------------------------------------------------------------------------
// Per-hop kernels
// ---------------------------------------------------------------------------
// vhop[j] = edge value after sparse edge-dropout (0 if dropped, else 2*val).
// JAX uniform pairs counters (p, p+HALF_E); keep iff u >= 0.5 <=> top bit set.
__global__ __launch_bounds__(256) void build_vhop(const float* __restrict__ val_s,
                                                  const int* __restrict__ eidx_s,
                                                  float* __restrict__ vhop,
                                                  uint32_t k0, uint32_t k1) {
  int j = blockIdx.x * 256 + threadIdx.x;
  if (j >= NNZ) return;
  int e = eidx_s[j];
  int p = (e < HALF_E) ? e : e - HALF_E;
  uint32_t b0, b1;
  threefry2x32(k0, k1, (uint32_t)p, (uint32_t)(p + HALF_E), b0, b1);
  uint32_t bits = (e < HALF_E) ? b0 : b1;
  vhop[j] = (bits & 0x80000000u) ? val_s[j] * 2.0f : 0.0f;
}

// WMMA SpMM: block = 16 rows x 4 waves (one 16-col tile/wave). Per step,
// 4 CSR edges form A[16x4] (one-hot: A[m,k]=v_k iff edge k targets row m)
// and B[4x16] (gathered source tiles); C[16x16] += A*B on the XDL pipe.
// A/B f32 K=4 VGPR layout (ISA 7.12.2): k = 2*(lane>=16) + vgpr_index.
// D layout: VGPR i -> row i + 8*(lane>=16), col lane&15.
template <int HOP>
__global__ __launch_bounds__(128)
void hop_gather_wmma(const float* __restrict__ vhop, const int* __restrict__ col_s,
                     const int* __restrict__ rowptr,
                     const float* __restrict__ userE, const float* __restrict__ itemE,
                     float* __restrict__ out) {
  constexpr int CAP = 1024;
  __shared__ float s_v[CAP];
  __shared__ int   s_c[CAP];
  __shared__ int   s_lr[CAP];
  __shared__ int   s_rp[17];
  const int t  = threadIdx.x;
  const int r0 = blockIdx.x * 16;
  if (t < 17) s_rp[t] = rowptr[r0 + t];
  __syncthreads();
  const int jbeg = s_rp[0];
  const int cnt  = s_rp[16] - jbeg;
  const bool stage = (cnt <= CAP);
  if (stage) {
    for (int j = t; j < cnt; j += 128) {
      // CDNA5 async copy: global -> LDS, tracked by ASYNCcnt.
      uint32_t lv = (uint32_t)(uintptr_t)&s_v[j];
      uint64_t gv = (uint64_t)(uintptr_t)(vhop + jbeg + j);
      asm volatile("global_load_async_to_lds_b32 %0, %1, off" :: "v"(lv), "v"(gv) : "memory");
      uint32_t lc = (uint32_t)(uintptr_t)&s_c[j];
      uint64_t gc = (uint64_t)(uintptr_t)(col_s + jbeg + j);
      asm volatile("global_load_async_to_lds_b32 %0, %1, off" :: "v"(lc), "v"(gc) : "memory");
      // Local destination row of CSR slot j (depends only on rowptr window).
      int J = jbeg + j, lr = 0;
      if (J >= s_rp[lr + 8]) lr += 8;
      if (J >= s_rp[lr + 4]) lr += 4;
      if (J >= s_rp[lr + 2]) lr += 2;
      if (J >= s_rp[lr + 1]) lr += 1;
      s_lr[j] = lr;
    }
    asm volatile("s_wait_asynccnt 0" ::: "memory");
  }
  __syncthreads();

  const int tile = t >> 5;       // wave -> column tile
  const int lane = t & 31;
  const int half = lane >> 4;    // lane group (K split)
  const int mn   = lane & 15;    // A-row m / B-col n
  const int kb   = half * 2;     // K index of VGPR0; VGPR1 is kb+1

  v8f acc = {};
  const int nsteps = (cnt + 3) >> 2;
  for (int s = 0; s < nsteps; ++s) {
    float a[2], b[2];
    const int base = 4 * s;
#pragma unroll
    for (int q = 0; q < 2; ++q) {
      const int idx = base + kb + q;
      float av = 0.f; int lr = 0, c = 0;
      if (idx < cnt) {
        if (stage) { av = s_v[idx]; lr = s_lr[idx]; c = s_c[idx]; }
        else {
          av = vhop[jbeg + idx];
          c  = col_s[jbeg + idx];
          int J = jbeg + idx;
          if (J >= s_rp[lr + 8]) lr += 8;
          if (J >= s_rp[lr + 4]) lr += 4;
          if (J >= s_rp[lr + 2]) lr += 2;
          if (J >= s_rp[lr + 1]) lr += 1;
        }
      }
      a[q] = (lr == mn) ? av : 0.f;
      float bv = 0.f;
      if (av != 0.f) {   // dropped/padded edges: keep B exactly 0 (0*NaN -> NaN!)
        const float* src;
        if (HOP == 0) src = (c < N_USERS) ? (userE + c * EMB) : (itemE + (c - N_USERS) * EMB);
        else          src = out + c * OUT_STRIDE + (HOP - 1) * EMB;
        bv = src[tile * 16 + mn];
      }
      b[q] = bv;
    }
    v2f A = { a[0], a[1] };
    v2f B = { b[0], b[1] };
    acc = __builtin_amdgcn_wmma_f32_16x16x4_f32(false, A, false, B, (short)0, acc,
                                                false, false);
  }

  float* dst = out + (r0 + 8 * half) * OUT_STRIDE + HOP * EMB + tile * 16 + mn;
#pragma unroll
  for (int i = 0; i < 8; ++i) dst[i * OUT_STRIDE] = acc[i];
}

// In-place message dropout on the hop's output slice.
// keep iff u >= 0.1; kept values scaled by 1/0.9. Counter pairing (p, p+HALF_M).
__global__ __launch_bounds__(256) void mess_dropout(float* __restrict__ out, int hop,
                                                    uint32_t k0, uint32_t k1) {
  int p = blockIdx.x * 256 + threadIdx.x;
  if (p >= HALF_M) return;
  uint32_t b0, b1;
  threefry2x32(k0, k1, (uint32_t)p, (uint32_t)(p + HALF_M), b0, b1);
  const int j0 = p, j1 = p + HALF_M;
  const int i0 = (j0 >> 6) * OUT_STRIDE + hop * EMB + (j0 & 63);
  const int i1 = (j1 >> 6) * OUT_STRIDE + hop * EMB + (j1 & 63);
  float x0 = out[i0], x1 = out[i1];
  out[i0] = (bits_to_unit(b0) >= 0.1f) ? x0 * (1.0f / 0.9f) : 0.0f;
  out[i1] = (bits_to_unit(b1) >= 0.1f) ? x1 * (1.0f / 0.9f) : 0.0f;
}

// ---------------------------------------------------------------------------
// Fallback (workspace too small): edge-parallel scatter with f32 atomics.
// ---------------------------------------------------------------------------
__global__ __launch_bounds__(256) void hop_scatter_atomic(const int* __restrict__ rows,
                                                          const int* __restrict__ cols,
                                                          const float* __restrict__ vals,
                                                          const float* __restrict__ userE,
                                                          const float* __restrict__ itemE,
                                                          float* __restrict__ out, int hop,
                                                          uint32_t k0, uint32_t k1) {
  int tid  = blockIdx.x * 256 + threadIdx.x;
  int pair = tid >> 5;
  int lane = tid & 31;
  if (pair >= HALF_E) return;
  uint32_t b0, b1;
  threefry2x32(k0, k1, (uint32_t)pair, (uint32_t)(pair + HALF_E), b0, b1);
  int half = lane >> 4, sub = lane & 15;
  uint32_t bits = half ? b1 : b0;
  if (!(bits & 0x80000000u)) return;   // dropped edge
  int e = pair + half * HALF_E;
  float v = vals[e] * 2.0f;
  int c = cols[e], r = rows[e];
  const float* src = (hop == 0)
      ? ((c < N_USERS) ? userE + c * EMB : itemE + (c - N_USERS) * EMB)
      : out + c * OUT_STRIDE + (hop - 1) * EMB;
  float4 x = ((const float4*)src)[sub];
  float* dst = out + r * OUT_STRIDE + hop * EMB + sub * 4;
  atomicAdd(dst + 0, v * x.x);
  atomicAdd(dst + 1, v * x.y);
  atomicAdd(dst + 2, v * x.z);
  atomicAdd(dst + 3, v * x.w);
}

// ---------------------------------------------------------------------------
// Host launcher
// ---------------------------------------------------------------------------
extern "C" void kernel_launch(void* const* d_in, const int* in_sizes, int n_in,
                              void* d_out, int out_size, void* d_ws, size_t ws_size,
                              hipStream_t stream) {
  (void)in_sizes; (void)n_in; (void)out_size;
  const float* userE = (const float*)d_in[0];
  const float* itemE = (const float*)d_in[1];
  const float* vals  = (const float*)d_in[2];
  const int*   rows  = (const int*)d_in[3];
  const int*   cols  = (const int*)d_in[4];
  float* out = (float*)d_out;

  // Hop keys: drop_key = key(42); ekey = fold_in(key, 2h); mkey = fold_in(key, 2h+1)
  uint32_t ek0[N_HOPS], ek1[N_HOPS], mk0[N_HOPS], mk1[N_HOPS];
  for (int h = 0; h < N_HOPS; ++h) {
    threefry2x32(0u, 42u, 0u, (uint32_t)(2 * h),     ek0[h], ek1[h]);
    threefry2x32(0u, 42u, 0u, (uint32_t)(2 * h + 1), mk0[h], mk1[h]);
  }

  // Workspace layout
  size_t off = 0;
  auto alloc = [&](size_t bytes) { size_t o = off; off = (off + bytes + 255) & ~(size_t)255; return o; };
  size_t o_cnt  = alloc((size_t)N_NODES * 4);
  size_t o_rowp = alloc((size_t)(N_NODES + 1) * 4);
  size_t o_cur  = alloc((size_t)N_NODES * 4);
  size_t o_bsum = alloc((size_t)SCAN_B * 4);
  size_t o_eidx = alloc((size_t)NNZ * 4);
  size_t o_col  = alloc((size_t)NNZ * 4);
  size_t o_val  = alloc((size_t)NNZ * 4);
  size_t o_vhop = alloc((size_t)NNZ * 4);
  size_t need = off;

  if (d_ws != nullptr && ws_size >= need) {
    char* w = (char*)d_ws;
    int*   cnt    = (int*)(w + o_cnt);
    int*   rowptr = (int*)(w + o_rowp);
    int*   cursor = (int*)(w + o_cur);
    int*   bsum   = (int*)(w + o_bsum);
    int*   eidx_s = (int*)(w + o_eidx);
    int*   col_s  = (int*)(w + o_col);
    float* val_s  = (float*)(w + o_val);
    float* vhop   = (float*)(w + o_vhop);

    zero_i32<<<(N_NODES + 255) / 256, 256, 0, stream>>>(cnt, N_NODES);
    hist_rows<<<(NNZ + 255) / 256, 256, 0, stream>>>(rows, cnt);
    scan_k1<<<SCAN_B, 256, 0, stream>>>(cnt, bsum);
    scan_k2<<<1, 32, 0, stream>>>(bsum, rowptr);
    scan_k3<<<SCAN_B, 256, 0, stream>>>(cnt, bsum, rowptr, cursor);
    scatter_csr<<<(NNZ + 255) / 256, 256, 0, stream>>>(rows, cols, vals, cursor,
                                                       eidx_s, col_s, val_s);
    sort_rows<<<(N_NODES + 255) / 256, 256, 0, stream>>>(rowptr, eidx_s, col_s, val_s);

    for (int h = 0; h < N_HOPS; ++h) {
      build_vhop<<<(NNZ + 255) / 256, 256, 0, stream>>>(val_s, eidx_s, vhop, ek0[h], ek1[h]);
      const int gblocks = N_NODES / 16;   // 18750
      if (h == 0)      hop_gather_wmma<0><<<gblocks, 128, 0, stream>>>(vhop, col_s, rowptr, userE, itemE, out);
      else if (h == 1) hop_gather_wmma<1><<<gblocks, 128, 0, stream>>>(vhop, col_s, rowptr, userE, itemE, out);
      else             hop_gather_wmma<2><<<gblocks, 128, 0, stream>>>(vhop, col_s, rowptr, userE, itemE, out);
      mess_dropout<<<(HALF_M + 255) / 256, 256, 0, stream>>>(out, h, mk0[h], mk1[h]);
    }
  } else {
    // Atomic fallback: zero the whole output once, scatter with atomics per hop.
    const int n4 = (N_NODES * OUT_STRIDE) / 4;
    zero_f4<<<(n4 + 255) / 256, 256, 0, stream>>>((float4*)out, n4);
    for (int h = 0; h < N_HOPS; ++h) {
      int threads = HALF_E * 32;
      hop_scatter_atomic<<<(threads + 255) / 256, 256, 0, stream>>>(
          rows, cols, vals, userE, itemE, out, h, ek0[h], ek1[h]);
      mess_dropout<<<(HALF_M + 255) / 256, 256, 0, stream>>>(out, h, mk0[h], mk1[h]);
    }
  }
}